// GNNModelObj_3925600108677
// MI455X (gfx1250) — compile-verified
//
#include <hip/hip_runtime.h>

typedef float v2f __attribute__((ext_vector_type(2)));
typedef float v8f __attribute__((ext_vector_type(8)));

#define HID 64

// ---------------- degree kernels ----------------

__global__ void init_deg_kernel(float* __restrict__ deg, int n) {
    int tid = blockIdx.x * blockDim.x + threadIdx.x;
    if (tid < n) deg[tid] = 1.0f;  // self-loop weight
}

__global__ void accum_deg_kernel(const long long* __restrict__ col,
                                 const float* __restrict__ ew,
                                 float* __restrict__ deg, int E) {
    int tid = blockIdx.x * blockDim.x + threadIdx.x;
    if (tid < E) atomicAdd(&deg[(int)col[tid]], ew[tid]);
}

__global__ void rsqrt_deg_kernel(float* __restrict__ deg, int n) {
    int tid = blockIdx.x * blockDim.x + threadIdx.x;
    if (tid < n) {
        float d = deg[tid];
        deg[tid] = (d > 0.0f) ? rsqrtf(d) : 0.0f;  // now holds dinv
    }
}

// ---------------- GEMM + row-scale: xs = (in @ W) * dinv[row]; acc = xs ----------------
// One wave computes a 16-row x 64-col tile with V_WMMA_F32_16X16X4_F32.
// A (16xK f32) layout: lane = m (0..15) | m+16 ; VGPR pair = K=k0, k0+1 with k0 = 2*(lane>>4).
// B (Kx16 f32) layout: lane = n | n+16 ; pair = rows k0, k0+1.
// C/D (16x16 f32): c[r] holds M = r + 8*(lane>>4), N = lane&15.

template <int K>
__global__ __launch_bounds__(256) void gcn_gemm_scale_kernel(
    const float* __restrict__ in, const float* __restrict__ W,
    const float* __restrict__ dinv, float* __restrict__ xs,
    float* __restrict__ acc, int n) {
    __shared__ float Ws[K * HID];
    for (int i = threadIdx.x; i < K * HID; i += 256) Ws[i] = W[i];
    __syncthreads();

#if defined(__gfx1250__) && __has_builtin(__builtin_amdgcn_wmma_f32_16x16x4_f32)
    const int lane = threadIdx.x & 31;
    const int wave = threadIdx.x >> 5;
    const int tile = blockIdx.x * 8 + wave;
    const int base = tile * 16;
    if (base >= n) return;  // wave-uniform: EXEC stays all-ones for WMMA

    const int m16 = lane & 15;
    const int hi = lane >> 4;
    const int k0 = hi * 2;
    const int arow = (base + m16 < n) ? (base + m16) : (n - 1);

    v8f c[4] = {};
#pragma unroll
    for (int ks = 0; ks < K / 4; ++ks) {
        const int kk = ks * 4 + k0;
        v2f a;
        a.x = in[arow * K + kk];
        a.y = in[arow * K + kk + 1];
#pragma unroll
        for (int t = 0; t < 4; ++t) {
            v2f b;
            b.x = Ws[kk * HID + t * 16 + m16];
            b.y = Ws[(kk + 1) * HID + t * 16 + m16];
            c[t] = __builtin_amdgcn_wmma_f32_16x16x4_f32(
                false, a, false, b, (short)0, c[t], false, false);
        }
    }
#pragma unroll
    for (int r = 0; r < 8; ++r) {
        const int m = base + r + 8 * hi;
        if (m < n) {
            const float dv = dinv[m];
#pragma unroll
            for (int t = 0; t < 4; ++t) {
                const float v = c[t][r] * dv;
                xs[m * HID + t * 16 + m16] = v;
                acc[m * HID + t * 16 + m16] = v;
            }
        }
    }
#else
    // scalar fallback (kept valid for host parse / non-wmma targets)
    const int rowBase = blockIdx.x * 128;
    for (int idx = threadIdx.x; idx < 128 * HID; idx += 256) {
        const int r = rowBase + (idx >> 6);
        if (r >= n) continue;
        const int f = idx & 63;
        float s = 0.0f;
        for (int k = 0; k < K; ++k) s += in[r * K + k] * Ws[k * HID + f];
        s *= dinv[r];
        xs[r * HID + f] = s;
        acc[r * HID + f] = s;
    }
#endif
}

// ---------------- edge scatter: acc[col] += xs[row] * w ----------------
// 64 threads per edge: coalesced 256B gather, coalesced f32 atomics (L2-resident).

__global__ __launch_bounds__(256) void scatter_edges_kernel(
    const long long* __restrict__ row, const long long* __restrict__ col,
    const float* __restrict__ ew, const float* __restrict__ xs,
    float* __restrict__ acc, int E) {
    int tid = blockIdx.x * 256 + threadIdx.x;
    int e = tid >> 6;
    if (e >= E) return;
    int f = tid & 63;
    int r = (int)row[e];
    int c = (int)col[e];
    float w = ew[e];
    atomicAdd(&acc[c * HID + f], xs[r * HID + f] * w);
}

// ---------------- finalize: out = relu(acc * dinv[row] + b) ----------------

__global__ void finalize_kernel(const float* __restrict__ acc,
                                const float* __restrict__ dinv,
                                const float* __restrict__ b,
                                float* __restrict__ out, int n) {
    int tid = blockIdx.x * blockDim.x + threadIdx.x;
    if (tid >= n * HID) return;
    int i = tid >> 6;
    int f = tid & 63;
    float v = acc[tid] * dinv[i] + b[f];
    out[tid] = (v > 0.0f) ? v : 0.0f;
}

extern "C" void kernel_launch(void* const* d_in, const int* in_sizes, int n_in,
                              void* d_out, int out_size, void* d_ws, size_t ws_size,
                              hipStream_t stream) {
    const float* x = (const float*)d_in[0];
    const long long* ei = (const long long*)d_in[1];  // int64 [2, E]
    const float* ew = (const float*)d_in[2];
    const float* W1 = (const float*)d_in[3];
    const float* b1 = (const float*)d_in[4];
    const float* W2 = (const float*)d_in[5];
    const float* b2 = (const float*)d_in[6];

    const int N = in_sizes[0] / 4;
    const int E = in_sizes[2];
    const long long* row = ei;
    const long long* col = ei + E;

    float* dinv = (float*)d_ws;                 // N floats
    float* xs = dinv + N;                       // N*64 floats
    float* acc = xs + (size_t)N * HID;          // N*64 floats
    float* h = (float*)d_out;                   // layer-1 activations staged in d_out
    float* out = (float*)d_out;

    const int nt = 256;
    const int nBlk = (N + nt - 1) / nt;
    const int eBlk = (E + nt - 1) / nt;
    const int tiles = (N + 15) / 16;
    const int gBlk = (tiles + 7) / 8;           // 8 waves (tiles) per block
    const int sBlk = (int)(((long long)E * HID + nt - 1) / nt);
    const int fBlk = (int)(((long long)N * HID + nt - 1) / nt);

    // degrees -> dinv (shared by both layers)
    init_deg_kernel<<<nBlk, nt, 0, stream>>>(dinv, N);
    accum_deg_kernel<<<eBlk, nt, 0, stream>>>(col, ew, dinv, E);
    rsqrt_deg_kernel<<<nBlk, nt, 0, stream>>>(dinv, N);

    // layer 1 (K = 4)
    gcn_gemm_scale_kernel<4><<<gBlk, nt, 0, stream>>>(x, W1, dinv, xs, acc, N);
    scatter_edges_kernel<<<sBlk, nt, 0, stream>>>(row, col, ew, xs, acc, E);
    finalize_kernel<<<fBlk, nt, 0, stream>>>(acc, dinv, b1, h, N);

    // layer 2 (K = 64)
    gcn_gemm_scale_kernel<64><<<gBlk, nt, 0, stream>>>(h, W2, dinv, xs, acc, N);
    scatter_edges_kernel<<<sBlk, nt, 0, stream>>>(row, col, ew, xs, acc, E);
    finalize_kernel<<<fBlk, nt, 0, stream>>>(acc, dinv, b2, out, N);
}